// CapsuleConv2d_34557306864261
// MI455X (gfx1250) — compile-verified
//
#include <hip/hip_runtime.h>
#include <hip/hip_bf16.h>

typedef float v2f __attribute__((ext_vector_type(2)));
typedef float v8f __attribute__((ext_vector_type(8)));

#define NIN 72          // G*KH*KW = 8*9 routing inputs
#define PRI_PITCH 260   // 256 + 4 pad -> lane bank spread
#define LDS_FLOATS (9*16*16 + NIN*PRI_PITCH + 256 + 256 + NIN*16)
#define N_TILES 144     // 9 ij * 16 ol-tiles
#define PW_ELEMS (N_TILES * 4 * 32 * 2)   // v2f per (tile, kstep, lane)

// ---- Repack weights into per-lane WMMA A-fragment order (coalesced b64 loads) ----
// packedW[((T*4 + ks)*32 + lane)] = { W_ij[ol0+ (lane&15)][ka], W_ij[...][ka+1] }
__global__ __launch_bounds__(256) void repack_weights(
    const float* __restrict__ w, float* __restrict__ pw)
{
  int idx  = blockIdx.x * 256 + threadIdx.x;   // [0, 144*4*32)
  int lane = idx & 31;
  int ks   = (idx >> 5) & 3;
  int T    = idx >> 7;
  int ij   = T >> 4;
  int ol0  = (T & 15) << 4;
  int row  = ol0 + (lane & 15);
  int ka   = (ks << 2) + ((lane >> 4) << 1);
  v2f a;
  a.x = w[row * 144 + ka * 9 + ij];
  a.y = w[row * 144 + (ka + 1) * 9 + ij];
  ((v2f*)pw)[idx] = a;
}

// One workgroup per output pixel (b,h,w). 256 threads = 8 wave32.
template <bool PACKED>
__global__ __launch_bounds__(256) void capsconv_route_kernel(
    const float* __restrict__ x,      // [2,128,32,32]
    const float* __restrict__ w,      // [16,16,16,3,3] -> W_ij[ol][m] = w[ol*144 + m*9 + ij]
    const float* __restrict__ pw,     // repacked A fragments (when PACKED)
    float* __restrict__ out)          // [2,256,32,32]
{
  extern __shared__ float lds[];
  float* patchT = lds;                          // [9][16][16]  (ij, m, g) g padded 8->16 w/ zeros
  float* priors = lds + 9*16*16;                // [72][260]    (n, ol)
  float* v_lds  = priors + NIN*PRI_PITCH;       // [256]        (o*16+l)
  float* vn_lds = v_lds + 256;                  // [256]
  float* lg_lds = vn_lds + 256;                 // [72][16]     (n, o)

  const int t   = threadIdx.x;
  const int bid = blockIdx.x;
  const int b    = bid >> 10;
  const int h    = (bid >> 5) & 31;
  const int wpix = bid & 31;

  // ---- Stage 1: stage 3x3x128 halo patch into LDS in WMMA-B layout ----
  #pragma unroll
  for (int k = 0; k < 9; ++k) {
    int idx = t + (k << 8);        // [0, 2304)
    int ij  = idx >> 8;
    int rem = idx & 255;
    int m   = rem >> 4;
    int g   = rem & 15;
    float val = 0.0f;
    if (g < 8) {
      int i = ij / 3, j = ij % 3;
      int hy = h + i - 1, wx = wpix + j - 1;
      if (hy >= 0 && hy < 32 && wx >= 0 && wx < 32) {
        int c = (g << 4) + m;
        val = x[((b * 128 + c) * 32 + hy) * 32 + wx];
      }
    }
    patchT[(ij * 16 + m) * 16 + g] = val;
  }
  __syncthreads();

  // ---- Stage 2: priors[n][ol] via V_WMMA_F32_16X16X4_F32 ----
  // Per tile T=(ij,olt): D[16x16] = W_ij[ol0:ol0+16, 0:16] x patchT[ij][0:16][g cols]
  const int wave = t >> 5;
  const int lane = t & 31;
  const int ln   = lane & 15;
  const int hi   = lane >> 4;
  #pragma unroll 2
  for (int k = 0; k < 18; ++k) {
    int T   = wave + (k << 3);     // [0,144)
    int ij  = T >> 4;
    int ol0 = (T & 15) << 4;
    v8f acc = {};
    #pragma unroll
    for (int ks = 0; ks < 4; ++ks) {
      int ka = (ks << 2) + (hi << 1);   // K index for this half-wave (ISA 16x4 A / 4x16 B layout)
      v2f a, bb;
      if (PACKED) {
        a = ((const v2f*)pw)[(T * 4 + ks) * 32 + lane];   // coalesced global_load_b64
      } else {
        int arow = ol0 + ln;
        a.x = w[arow * 144 + ka * 9 + ij];
        a.y = w[arow * 144 + (ka + 1) * 9 + ij];
      }
      bb.x = patchT[(ij * 16 + ka) * 16 + ln];
      bb.y = patchT[(ij * 16 + ka + 1) * 16 + ln];
      acc = __builtin_amdgcn_wmma_f32_16x16x4_f32(
          /*neg_a=*/false, a, /*neg_b=*/false, bb,
          /*c_mod=*/(short)0, acc, /*reuse_a=*/false, /*reuse_b=*/false);
    }
    // D: row = ol0 + vr + 8*hi, col = ln (= g, valid g<8)
    if (ln < 8) {
      int n = (ij << 3) + ln;
      float* dst = &priors[n * PRI_PITCH + ol0 + (hi << 3)];
      #pragma unroll
      for (int vr = 0; vr < 8; ++vr) dst[vr] = acc[vr];
    }
  }
  __syncthreads();

  // ---- Stage 3: k-means (dot) routing, thread t <-> (o,l), ol == t ----
  const int o = t >> 4;
  float v = 0.0f;
  for (int n = 0; n < NIN; ++n) v += priors[n * PRI_PITCH + t];
  v *= (1.0f / 72.0f);

  for (int it = 0; it < 3; ++it) {
    __syncthreads();
    v_lds[t] = v;
    __syncthreads();
    // v / max(||v||, eps) per capsule o (redundant across the 16 (o,*) threads)
    float s2 = 0.0f;
    const float* vo = &v_lds[o << 4];
    #pragma unroll
    for (int ll = 0; ll < 16; ++ll) { float vv = vo[ll]; s2 += vv * vv; }
    float denom = fmaxf(sqrtf(s2), 1e-12f);
    vn_lds[t] = v / denom;
    __syncthreads();
    // logits[n][o] = <u[n,o,:], vhat[o,:]>   (1152 items over 256 threads)
    #pragma unroll
    for (int k = 0; k < 5; ++k) {
      int item = t + (k << 8);
      if (item < NIN * 16) {
        int n  = item >> 4;
        int oo = item & 15;
        const float* up = &priors[n * PRI_PITCH + (oo << 4)];
        const float* vp = &vn_lds[oo << 4];
        float lg = 0.0f;
        #pragma unroll
        for (int ll = 0; ll < 16; ++ll) lg += up[ll] * vp[ll];
        lg_lds[item] = lg;
      }
    }
    __syncthreads();
    // softmax over n (per o) fused with weighted sum -> new v (this thread's (o,l))
    float mx = -3.402823466e+38f;
    for (int n = 0; n < NIN; ++n) mx = fmaxf(mx, lg_lds[(n << 4) + o]);
    float ssum = 0.0f, wsum = 0.0f;
    for (int n = 0; n < NIN; ++n) {
      float p = __expf(lg_lds[(n << 4) + o] - mx);
      ssum += p;
      wsum += p * priors[n * PRI_PITCH + t];
    }
    v = wsum / ssum;
  }

  // ---- Stage 4: squash + write [B, 256, 32, 32] ----
  __syncthreads();
  v_lds[t] = v;
  __syncthreads();
  float s2 = 0.0f;
  const float* vo = &v_lds[o << 4];
  #pragma unroll
  for (int ll = 0; ll < 16; ++ll) { float vv = vo[ll]; s2 += vv * vv; }
  float nrm = sqrtf(s2);
  out[((b * 256 + t) * 32 + h) * 32 + wpix] = v * (nrm / (1.0f + s2));
}

extern "C" void kernel_launch(void* const* d_in, const int* in_sizes, int n_in,
                              void* d_out, int out_size, void* d_ws, size_t ws_size,
                              hipStream_t stream) {
  const float* x = (const float*)d_in[0];   // [2,128,32,32] fp32
  const float* w = (const float*)d_in[1];   // [16,16,16,3,3] fp32
  float* out = (float*)d_out;               // [2,256,32,32] fp32
  (void)in_sizes; (void)n_in; (void)out_size;
  const size_t shmem = (size_t)LDS_FLOATS * sizeof(float);   // 90,752 B < 320 KB
  const size_t pw_bytes = (size_t)PW_ELEMS * sizeof(float);  // 147,456 B
  if (d_ws != nullptr && ws_size >= pw_bytes) {
    float* pw = (float*)d_ws;
    repack_weights<<<dim3(N_TILES * 4 * 32 / 256), dim3(256), 0, stream>>>(w, pw);
    capsconv_route_kernel<true><<<dim3(2048), dim3(256), shmem, stream>>>(x, w, pw, out);
  } else {
    capsconv_route_kernel<false><<<dim3(2048), dim3(256), shmem, stream>>>(x, w, nullptr, out);
  }
}